// DynamicConvBlock_43954695307645
// MI455X (gfx1250) — compile-verified
//
#include <hip/hip_runtime.h>
#include <math.h>

// Problem constants (match reference)
#define C_CH 256
#define HW   4096
#define B_N  16
#define E_N  4
#define CR   64
#define NPS  (C_CH * HW)   // elements per sample for GroupNorm

typedef __attribute__((ext_vector_type(2))) float v2f;
typedef __attribute__((ext_vector_type(8))) float v8f;
typedef __attribute__((ext_vector_type(4))) int   v4i;

#define GAS __attribute__((address_space(1)))   // global  (prints as __device__)
#define LAS __attribute__((address_space(3)))   // LDS     (prints as __shared__)

// ---------------------------------------------------------------------------
// Tile loader: 64x64 fp32 channel (16 KB) -> LDS.
// Preferred path: CDNA5 async global->LDS B128 (no VGPR round trip, ASYNCcnt).
// Each of 256 threads issues 4 x 16B transfers, all 16B-aligned.
// Caller must __syncthreads() after (each wave waits its own ASYNCcnt first).
// ---------------------------------------------------------------------------
__device__ __forceinline__ void load_tile(const float* __restrict__ xc, float* tile) {
#if __has_builtin(__builtin_amdgcn_global_load_async_to_lds_b128)
    const uintptr_t gbase = (uintptr_t)xc + (uintptr_t)threadIdx.x * 16u;
    const uint32_t  lbase = (uint32_t)(uintptr_t)tile + threadIdx.x * 16u;
    #pragma unroll
    for (int i = 0; i < 4; ++i) {
        __builtin_amdgcn_global_load_async_to_lds_b128(
            (GAS v4i*)(gbase + (uintptr_t)(i * 4096)),
            (LAS v4i*)(lbase + (uint32_t)(i * 4096)),
            0, 0);
    }
#if __has_builtin(__builtin_amdgcn_s_wait_asynccnt)
    __builtin_amdgcn_s_wait_asynccnt(0);
#else
    asm volatile("s_wait_asynccnt 0x0" ::: "memory");
#endif
#else
    #pragma unroll
    for (int i = 0; i < 16; ++i)
        tile[threadIdx.x + i * 256] = xc[threadIdx.x + i * 256];
#endif
}

// ---------------------------------------------------------------------------
// K1: global average pool per (b,c). grid = B*C blocks, 256 threads.
// x layout [b][c][h][w] => channel (b,c) is 4096 contiguous floats at bc*4096.
// ---------------------------------------------------------------------------
__global__ __launch_bounds__(256) void pool_kernel(const float* __restrict__ x,
                                                   float* __restrict__ pooled) {
    __shared__ float red[256];
    const int bc = blockIdx.x;
    const float* p = x + (size_t)bc * HW;
    float s = 0.f;
    #pragma unroll
    for (int i = 0; i < 16; ++i) s += p[threadIdx.x + i * 256];
    red[threadIdx.x] = s;
    __syncthreads();
    for (int off = 128; off > 0; off >>= 1) {
        if ((int)threadIdx.x < off) red[threadIdx.x] += red[threadIdx.x + off];
        __syncthreads();
    }
    if (threadIdx.x == 0) pooled[bc] = red[0] * (1.0f / HW);
}

// ---------------------------------------------------------------------------
// K2: gate MLP. One block, 128 threads = 4 waves (wave32).
// H1 = relu(pooled(16x256) @ w1^T(256x64) + b1) via V_WMMA_F32_16X16X4_F32:
//   wave n owns N-tile n (16 cols), 64 K-steps of 4.
// Then logits = H1 @ w2^T + b2 (tiny, VALU), softmax over E=4.
// ---------------------------------------------------------------------------
__global__ __launch_bounds__(128) void gate_kernel(const float* __restrict__ pooled,
                                                   const float* __restrict__ w1,
                                                   const float* __restrict__ b1,
                                                   const float* __restrict__ w2,
                                                   const float* __restrict__ b2,
                                                   float* __restrict__ a_out) {
    __shared__ float h_lds[16][64];
    __shared__ float logits[16][4];
    const int tid  = threadIdx.x;
    const int wave = tid >> 5;
    const int lane = tid & 31;
    const int m    = lane & 15;            // M row (A) / N col within tile (B)
    const int koff = (lane >> 4) << 1;     // lanes 0-15: K=k0,k0+1 ; 16-31: K=k0+2,k0+3
    const int n0   = wave << 4;            // N-tile base
    const int nrow = n0 + m;               // w1 row feeding this lane's B frag

    v8f acc = {};
    for (int k0 = 0; k0 < C_CH; k0 += 4) {
        v2f a, b;
        a.x = pooled[m * C_CH + k0 + koff];
        a.y = pooled[m * C_CH + k0 + koff + 1];
        b.x = w1[nrow * C_CH + k0 + koff];
        b.y = w1[nrow * C_CH + k0 + koff + 1];
        acc = __builtin_amdgcn_wmma_f32_16x16x4_f32(false, a, false, b,
                                                    (short)0, acc, false, false);
    }
    // D layout: VGPR r -> (lanes 0-15: M=r, N=lane) (lanes 16-31: M=r+8, N=lane-16)
    const int   col  = n0 + m;
    const float bias = b1[col];
    #pragma unroll
    for (int r = 0; r < 8; ++r) {
        const int row = (lane < 16) ? r : (r + 8);
        const float v = acc[r] + bias;
        h_lds[row][col] = v > 0.f ? v : 0.f;   // ReLU
    }
    __syncthreads();

    if (tid < 64) {                        // 16 batches x 4 experts
        const int b_ = tid >> 2, e = tid & 3;
        float s = b2[e];
        #pragma unroll 8
        for (int j = 0; j < CR; ++j) s += h_lds[b_][j] * w2[e * CR + j];
        logits[b_][e] = s;
    }
    __syncthreads();
    if (tid < 16) {
        const float l0 = logits[tid][0], l1 = logits[tid][1];
        const float l2 = logits[tid][2], l3 = logits[tid][3];
        const float mx = fmaxf(fmaxf(l0, l1), fmaxf(l2, l3));
        const float e0 = __expf(l0 - mx), e1 = __expf(l1 - mx);
        const float e2 = __expf(l2 - mx), e3 = __expf(l3 - mx);
        const float inv = 1.f / (e0 + e1 + e2 + e3);
        a_out[tid * 4 + 0] = e0 * inv;
        a_out[tid * 4 + 1] = e1 * inv;
        a_out[tid * 4 + 2] = e2 * inv;
        a_out[tid * 4 + 3] = e3 * inv;
    }
}

// ---------------------------------------------------------------------------
// K3: fold expert blend into weights:
//   wcomb[b][c][t] = sum_e a[b,e]*conv_w[e][c][t] ; bcomb[b][c] = sum_e a*conv_b
// grid = B blocks, 256 threads (one per channel).
// ---------------------------------------------------------------------------
__global__ __launch_bounds__(256) void combine_kernel(const float* __restrict__ a_gate,
                                                      const float* __restrict__ conv_w,
                                                      const float* __restrict__ conv_b,
                                                      float* __restrict__ wcomb,
                                                      float* __restrict__ bcomb) {
    const int b = blockIdx.x, c = threadIdx.x;
    const float a0 = a_gate[b * 4 + 0], a1 = a_gate[b * 4 + 1];
    const float a2 = a_gate[b * 4 + 2], a3 = a_gate[b * 4 + 3];
    #pragma unroll
    for (int t = 0; t < 9; ++t) {
        const float w = a0 * conv_w[(0 * C_CH + c) * 9 + t]
                      + a1 * conv_w[(1 * C_CH + c) * 9 + t]
                      + a2 * conv_w[(2 * C_CH + c) * 9 + t]
                      + a3 * conv_w[(3 * C_CH + c) * 9 + t];
        wcomb[(b * C_CH + c) * 9 + t] = w;
    }
    bcomb[b * C_CH + c] = a0 * conv_b[0 * C_CH + c] + a1 * conv_b[1 * C_CH + c]
                        + a2 * conv_b[2 * C_CH + c] + a3 * conv_b[3 * C_CH + c];
}

// ---------------------------------------------------------------------------
// Depthwise 3x3 at (h,w) with zero "SAME" padding (predicated edges).
// ---------------------------------------------------------------------------
__device__ __forceinline__ float conv_at(const float* tile, const float* wgt,
                                         int h, int w) {
    float acc = 0.f;
    #pragma unroll
    for (int dh = -1; dh <= 1; ++dh) {
        const int hh = h + dh;
        if ((unsigned)hh < 64u) {
            #pragma unroll
            for (int dw = -1; dw <= 1; ++dw) {
                const int ww = w + dw;
                if ((unsigned)ww < 64u)
                    acc += wgt[(dh + 1) * 3 + (dw + 1)] * tile[hh * 64 + ww];
            }
        }
    }
    return acc;
}

// ---------------------------------------------------------------------------
// K4: conv pass #1 -> per-channel partial sum / sumsq. grid = B*C, 256 thr.
// ---------------------------------------------------------------------------
__global__ __launch_bounds__(256) void stats_kernel(const float* __restrict__ x,
                                                    const float* __restrict__ wcomb,
                                                    const float* __restrict__ bcomb,
                                                    float* __restrict__ psum,
                                                    float* __restrict__ psq) {
    __shared__ float tile[HW];
    __shared__ float r1[256], r2[256];
    const int bc = blockIdx.x;
    load_tile(x + (size_t)bc * HW, tile);
    float wgt[9];
    #pragma unroll
    for (int t = 0; t < 9; ++t) wgt[t] = wcomb[bc * 9 + t];
    const float bias = bcomb[bc];
    __syncthreads();

    const int w0 = threadIdx.x & 63, h0 = threadIdx.x >> 6;
    float s = 0.f, sq = 0.f;
    #pragma unroll
    for (int i = 0; i < 16; ++i) {
        const float y = bias + conv_at(tile, wgt, h0 + i * 4, w0);
        s += y; sq += y * y;
    }
    r1[threadIdx.x] = s; r2[threadIdx.x] = sq;
    __syncthreads();
    for (int off = 128; off > 0; off >>= 1) {
        if ((int)threadIdx.x < off) {
            r1[threadIdx.x] += r1[threadIdx.x + off];
            r2[threadIdx.x] += r2[threadIdx.x + off];
        }
        __syncthreads();
    }
    if (threadIdx.x == 0) { psum[bc] = r1[0]; psq[bc] = r2[0]; }
}

// ---------------------------------------------------------------------------
// K5: reduce 256 channel-partials per sample (double, fixed order) -> mean,rstd
// ---------------------------------------------------------------------------
__global__ __launch_bounds__(256) void finalize_stats_kernel(const float* __restrict__ psum,
                                                             const float* __restrict__ psq,
                                                             float* __restrict__ mean_rstd) {
    __shared__ double r1[256], r2[256];
    const int b = blockIdx.x;
    r1[threadIdx.x] = (double)psum[b * C_CH + threadIdx.x];
    r2[threadIdx.x] = (double)psq[b * C_CH + threadIdx.x];
    __syncthreads();
    for (int off = 128; off > 0; off >>= 1) {
        if ((int)threadIdx.x < off) {
            r1[threadIdx.x] += r1[threadIdx.x + off];
            r2[threadIdx.x] += r2[threadIdx.x + off];
        }
        __syncthreads();
    }
    if (threadIdx.x == 0) {
        const double n = (double)NPS;
        const double mean = r1[0] / n;
        const double var  = r2[0] / n - mean * mean;
        mean_rstd[b * 2 + 0] = (float)mean;
        mean_rstd[b * 2 + 1] = (float)(1.0 / sqrt(var + 1e-5));
    }
}

// ---------------------------------------------------------------------------
// K6: conv pass #2 (recompute) -> GroupNorm affine -> SiLU -> out.
// ---------------------------------------------------------------------------
__global__ __launch_bounds__(256) void output_kernel(const float* __restrict__ x,
                                                     const float* __restrict__ wcomb,
                                                     const float* __restrict__ bcomb,
                                                     const float* __restrict__ mean_rstd,
                                                     const float* __restrict__ gamma,
                                                     const float* __restrict__ beta,
                                                     float* __restrict__ out) {
    __shared__ float tile[HW];
    const int bc = blockIdx.x;
    const int b  = bc >> 8, c = bc & 255;
    load_tile(x + (size_t)bc * HW, tile);
    float wgt[9];
    #pragma unroll
    for (int t = 0; t < 9; ++t) wgt[t] = wcomb[bc * 9 + t];
    const float bias = bcomb[bc];
    const float mu   = mean_rstd[b * 2 + 0];
    const float rstd = mean_rstd[b * 2 + 1];
    const float g = gamma[c], bt = beta[c];
    __syncthreads();

    const int w0 = threadIdx.x & 63, h0 = threadIdx.x >> 6;
    float* op = out + (size_t)bc * HW;
    #pragma unroll
    for (int i = 0; i < 16; ++i) {
        const int h = h0 + i * 4;
        const float y = bias + conv_at(tile, wgt, h, w0);
        const float z = (y - mu) * rstd * g + bt;
        op[h * 64 + w0] = z / (1.f + __expf(-z));   // SiLU
    }
}

// ---------------------------------------------------------------------------
extern "C" void kernel_launch(void* const* d_in, const int* in_sizes, int n_in,
                              void* d_out, int out_size, void* d_ws, size_t ws_size,
                              hipStream_t stream) {
    const float* x      = (const float*)d_in[0];
    const float* w1     = (const float*)d_in[1];
    const float* b1     = (const float*)d_in[2];
    const float* w2     = (const float*)d_in[3];
    const float* b2     = (const float*)d_in[4];
    const float* conv_w = (const float*)d_in[5];
    const float* conv_b = (const float*)d_in[6];
    const float* gamma  = (const float*)d_in[7];
    const float* beta   = (const float*)d_in[8];
    float* out = (float*)d_out;
    float* ws  = (float*)d_ws;

    // workspace layout (floats): ~213 KB total
    float* pooled = ws;             // 4096
    float* a_gate = ws + 4096;      // 64
    float* wcomb  = ws + 4160;      // 36864
    float* bcomb  = ws + 41024;     // 4096
    float* psum   = ws + 45120;     // 4096
    float* psq    = ws + 49216;     // 4096
    float* mrs    = ws + 53312;     // 32  (8B aligned)

    pool_kernel          <<<B_N * C_CH, 256, 0, stream>>>(x, pooled);
    gate_kernel          <<<1,          128, 0, stream>>>(pooled, w1, b1, w2, b2, a_gate);
    combine_kernel       <<<B_N,        256, 0, stream>>>(a_gate, conv_w, conv_b, wcomb, bcomb);
    stats_kernel         <<<B_N * C_CH, 256, 0, stream>>>(x, wcomb, bcomb, psum, psq);
    finalize_stats_kernel<<<B_N,        256, 0, stream>>>(psum, psq, mrs);
    output_kernel        <<<B_N * C_CH, 256, 0, stream>>>(x, wcomb, bcomb, mrs, gamma, beta, out);
}